// AUVRNNStepModel_50096498541343
// MI455X (gfx1250) — compile-verified
//
#include <hip/hip_runtime.h>
#include <math.h>

typedef __attribute__((ext_vector_type(2))) float v2f;
typedef __attribute__((ext_vector_type(4))) float v4f;
typedef __attribute__((ext_vector_type(8))) float v8f;

#define K_TOT 262144
#define DT_C 0.1f
#define HA_S 130   // per-wave activation staging stride
#define W2_S 130   // W2 stride in LDS
#define W3_S 130   // padded W3 (16 rows, rows 6..15 zero)
#define DV_S 16    // dv staging stride
#define XS_S 38    // xc staging stride (K=36 padded)
#define H1_S 18    // h1 staging stride (16 cols, 12 valid)

__device__ __forceinline__ float lrelu(float z) { return fmaxf(z, 0.1f * z); }

__device__ __forceinline__ void mat3mul(float* __restrict__ d,
                                        const float* __restrict__ a,
                                        const float* __restrict__ b) {
  #pragma unroll
  for (int i = 0; i < 3; ++i)
    #pragma unroll
    for (int j = 0; j < 3; ++j)
      d[i*3+j] = a[i*3+0]*b[0*3+j] + a[i*3+1]*b[1*3+j] + a[i*3+2]*b[2*3+j];
}

__device__ __forceinline__ void matvec3(float* __restrict__ d,
                                        const float* __restrict__ m,
                                        const float* __restrict__ v) {
  #pragma unroll
  for (int i = 0; i < 3; ++i)
    d[i] = m[i*3+0]*v[0] + m[i*3+1]*v[1] + m[i*3+2]*v[2];
}

__global__ __launch_bounds__(128) void auv_step_kernel(
    const float* __restrict__ g_s,  const float* __restrict__ g_a,
    const float* __restrict__ g_h0, const float* __restrict__ g_Wih,
    const float* __restrict__ g_Whh, const float* __restrict__ g_W1,
    const float* __restrict__ g_W2, const float* __restrict__ g_W3,
    float* __restrict__ g_out)
{
  __shared__ __attribute__((aligned(16))) float W2s[128 * W2_S];    // 66560 B
  __shared__ __attribute__((aligned(16))) float HA[4 * 16 * HA_S];  // 33280 B
  __shared__ __attribute__((aligned(16))) float Xs[128 * XS_S];     // 19456 B
  __shared__ __attribute__((aligned(16))) float H1s[128 * H1_S];    // 9216 B
  __shared__ __attribute__((aligned(16))) float DVs[128 * DV_S];    // 8192 B
  __shared__ __attribute__((aligned(16))) float W3s[16 * W3_S];     // 8320 B
  __shared__ __attribute__((aligned(16))) float W1s[128 * 12];      // 6144 B
  __shared__ __attribute__((aligned(16))) float Wcomb[16 * XS_S];   // 2432 B
  // total = 153600 B (<=160KB -> 2 workgroups/WGP)

  const int tid  = threadIdx.x;
  const int wave = tid >> 5;
  const int lane = tid & 31;
  const int half = lane >> 4;   // upper half-wave carries K+2,K+3 of each K=4 step
  const int l16  = lane & 15;
  const int row  = tid;         // block-local row 0..127
  const int item = blockIdx.x * 128 + row;

  // ---- cooperative weight staging ----
  for (int i = tid; i < 128 * 128; i += 128)
    W2s[(i >> 7) * W2_S + (i & 127)] = g_W2[i];
  for (int i = tid; i < 128 * 12; i += 128) W1s[i] = g_W1[i];
  for (int i = tid; i < 16 * W3_S; i += 128) {
    const int r = i / W3_S, c = i - r * W3_S;
    W3s[i] = (r < 6 && c < 128) ? g_W3[r * 128 + c] : 0.f;
  }
  // Wcomb[n][k]: k<21 -> W_ih[n][k]; 21<=k<33 -> W_hh[n][k-21]; else 0; rows n>=12 zero
  for (int i = tid; i < 16 * XS_S; i += 128) {
    const int n = i / XS_S, k = i - n * XS_S;
    float v = 0.f;
    if (n < 12) {
      if (k < 21)      v = g_Wih[n * 21 + k];
      else if (k < 33) v = g_Whh[n * 12 + (k - 21)];
    }
    Wcomb[i] = v;
  }
  __syncthreads();

  // ---- Phase A: build xc = [x(21) | h0(12) | 0,0,0], stage to LDS (wave-private rows)
  {
    float xc[36];
    const v2f* sp = (const v2f*)(g_s + (size_t)item * 18);
    float s18[18];
    #pragma unroll
    for (int j = 0; j < 9; ++j) { v2f t = sp[j]; s18[2*j] = t.x; s18[2*j+1] = t.y; }
    #pragma unroll
    for (int j = 0; j < 15; ++j) xc[j] = s18[3 + j];
    const v2f* ap = (const v2f*)(g_a + (size_t)item * 6);
    #pragma unroll
    for (int j = 0; j < 3; ++j) { v2f t = ap[j]; xc[15+2*j] = t.x; xc[16+2*j] = t.y; }
    const v2f* hp = (const v2f*)(g_h0 + (size_t)item * 12);
    #pragma unroll
    for (int j = 0; j < 6; ++j) { v2f t = hp[j]; xc[21+2*j] = t.x; xc[22+2*j] = t.y; }
    xc[33] = 0.f; xc[34] = 0.f; xc[35] = 0.f;
    #pragma unroll
    for (int j = 0; j < 18; ++j)
      *(v2f*)&Xs[row * XS_S + 2 * j] = v2f{xc[2*j], xc[2*j+1]};
  }
  asm volatile("s_wait_dscnt 0x0" ::: "memory");

  // ---- GEMM phases: each wave owns rows [32w, 32w+32), two 16-row slabs ----
  float* HAw = &HA[wave * 16 * HA_S];
  float* outHbase = g_out + (size_t)K_TOT * 18;

  for (int slab = 0; slab < 2; ++slab) {
    const int srow = wave * 32 + slab * 16;

    // GEMM0: h1pre(16x16, 12 valid) = xc(16x36) @ Wcomb^T, 9 K-steps x 1 N-tile
    {
      v8f acc = {0.f, 0.f, 0.f, 0.f, 0.f, 0.f, 0.f, 0.f};
      #pragma unroll
      for (int j = 0; j < 9; ++j) {
        const int kb = 4 * j + 2 * half;
        v2f A = *(const v2f*)&Xs[(srow + l16) * XS_S + kb];
        v2f B = *(const v2f*)&Wcomb[l16 * XS_S + kb];
        acc = __builtin_amdgcn_wmma_f32_16x16x4_f32(
            false, A, false, B, (short)0, acc, false, false);
      }
      // tanh on D-layout regs; stage h1 + write hN output (cols 0..11 valid)
      #pragma unroll
      for (int v = 0; v < 8; ++v) {
        const int m = srow + v + 8 * half;
        float t = tanhf(acc[v]);
        H1s[m * H1_S + l16] = t;
        if (l16 < 12)
          outHbase[(size_t)(blockIdx.x * 128 + m) * 12 + l16] = t;
      }
    }
    asm volatile("s_wait_dscnt 0x0" ::: "memory");

    // GEMM1: h_a(16x128) = lrelu( h1(16x12) @ W1^T ), 3 K-steps x 8 N-tiles
    {
      v8f acc[8] = {};
      #pragma unroll
      for (int j = 0; j < 3; ++j) {
        const int kb = 4 * j + 2 * half;
        v2f A = *(const v2f*)&H1s[(srow + l16) * H1_S + kb];
        #pragma unroll
        for (int t = 0; t < 8; ++t) {
          v2f B = *(const v2f*)&W1s[(t * 16 + l16) * 12 + kb];
          acc[t] = __builtin_amdgcn_wmma_f32_16x16x4_f32(
              false, A, false, B, (short)0, acc[t], false, false);
        }
      }
      #pragma unroll
      for (int t = 0; t < 8; ++t)
        #pragma unroll
        for (int v = 0; v < 8; ++v)
          HAw[(v + 8 * half) * HA_S + t * 16 + l16] = lrelu(acc[t][v]);
    }
    asm volatile("s_wait_dscnt 0x0" ::: "memory");

    // GEMM2: h2(16x128) = lrelu( h_a @ W2^T ), 32 K-steps x 8 N-tiles
    {
      v8f acc[8] = {};
      for (int j = 0; j < 32; ++j) {
        const int kb = 4 * j + 2 * half;
        v2f A = *(const v2f*)&HAw[l16 * HA_S + kb];
        #pragma unroll
        for (int t = 0; t < 8; ++t) {
          v2f B = *(const v2f*)&W2s[(t * 16 + l16) * W2_S + kb];
          acc[t] = __builtin_amdgcn_wmma_f32_16x16x4_f32(
              false, A, false, B, (short)0, acc[t], false, false);
        }
      }
      asm volatile("s_wait_dscnt 0x0" ::: "memory");
      #pragma unroll
      for (int t = 0; t < 8; ++t)
        #pragma unroll
        for (int v = 0; v < 8; ++v)
          HAw[(v + 8 * half) * HA_S + t * 16 + l16] = lrelu(acc[t][v]);
    }
    asm volatile("s_wait_dscnt 0x0" ::: "memory");

    // GEMM3: dv(16x6) = h2(16x128) @ W3^T (padded to 16 cols), 32 K-steps x 1 N-tile
    {
      v8f acc = {0.f, 0.f, 0.f, 0.f, 0.f, 0.f, 0.f, 0.f};
      for (int j = 0; j < 32; ++j) {
        const int kb = 4 * j + 2 * half;
        v2f A = *(const v2f*)&HAw[l16 * HA_S + kb];
        v2f B = *(const v2f*)&W3s[l16 * W3_S + kb];
        acc = __builtin_amdgcn_wmma_f32_16x16x4_f32(
            false, A, false, B, (short)0, acc, false, false);
      }
      #pragma unroll
      for (int v = 0; v < 8; ++v)
        DVs[(srow + v + 8 * half) * DV_S + l16] = acc[v];
    }
    asm volatile("s_wait_dscnt 0x0" ::: "memory");
  }

  // ---- gather dv for this lane's item ----
  float dv[6];
  #pragma unroll
  for (int j = 0; j < 3; ++j) {
    v2f t = *(const v2f*)&DVs[row * DV_S + 2 * j];
    dv[2*j] = t.x; dv[2*j+1] = t.y;
  }

  // ---- SE(3) geometry per lane ----
  {
    float s18[18];
    const v2f* sp = (const v2f*)(g_s + (size_t)item * 18);
    #pragma unroll
    for (int j = 0; j < 9; ++j) { v2f t = sp[j]; s18[2*j] = t.x; s18[2*j+1] = t.y; }
    float* P  = &s18[0];
    float* R  = &s18[3];
    float* V6 = &s18[12];

    float rho[3] = {V6[0] * DT_C, V6[1] * DT_C, V6[2] * DT_C};
    float ph[3]  = {V6[3] * DT_C, V6[4] * DT_C, V6[5] * DT_C};
    float th2 = ph[0]*ph[0] + ph[1]*ph[1] + ph[2]*ph[2];
    const bool sm = th2 < 1e-8f;
    float th2s = sm ? 1.f : th2;
    float th = sqrtf(th2s);
    float sth = sinf(th), cth = cosf(th);
    float cA = sm ? (1.f - th2 * (1.f / 6.f))   : sth / th;
    float cB = sm ? (0.5f - th2 * (1.f / 24.f)) : (1.f - cth) / th2s;
    float cC = sm ? ((1.f / 6.f) - th2 * (1.f / 120.f)) : (th - sth) / (th2s * th);

    float S[9]  = {0.f, -ph[2], ph[1],  ph[2], 0.f, -ph[0],  -ph[1], ph[0], 0.f};
    float S2[9];
    mat3mul(S2, S, S);
    float Re[9], Vm[9];
    #pragma unroll
    for (int i = 0; i < 9; ++i) {
      float id = (i % 4 == 0) ? 1.f : 0.f;
      Re[i] = id + cA * S[i] + cB * S2[i];
      Vm[i] = id + cB * S[i] + cC * S2[i];
    }
    float pe[3]; matvec3(pe, Vm, rho);
    float Rn[9]; mat3mul(Rn, R, Re);
    float pn[3]; matvec3(pn, R, pe);
    #pragma unroll
    for (int i = 0; i < 3; ++i) pn[i] += P[i];

    float Ri[9] = {Rn[0], Rn[3], Rn[6],  Rn[1], Rn[4], Rn[7],  Rn[2], Rn[5], Rn[8]};
    float pinv[3]; matvec3(pinv, Ri, pn);
    #pragma unroll
    for (int i = 0; i < 3; ++i) pinv[i] = -pinv[i];

    // w = Adjoint(R, p) @ v + dv
    float Rw[3]; matvec3(Rw, R, &V6[3]);
    float Rv[3]; matvec3(Rv, R, &V6[0]);
    float w6[6];
    w6[0] = Rv[0] + (P[1]*Rw[2] - P[2]*Rw[1]) + dv[0];
    w6[1] = Rv[1] + (P[2]*Rw[0] - P[0]*Rw[2]) + dv[1];
    w6[2] = Rv[2] + (P[0]*Rw[1] - P[1]*Rw[0]) + dv[2];
    w6[3] = Rw[0] + dv[3];
    w6[4] = Rw[1] + dv[4];
    w6[5] = Rw[2] + dv[5];

    // v_next = Adjoint(Rinv, pinv) @ w
    float Ra[3]; matvec3(Ra, Ri, &w6[3]);
    float Rl[3]; matvec3(Rl, Ri, &w6[0]);

    float o18[18];
    o18[0] = pn[0]; o18[1] = pn[1]; o18[2] = pn[2];
    #pragma unroll
    for (int i = 0; i < 9; ++i) o18[3 + i] = Rn[i];
    o18[12] = Rl[0] + (pinv[1]*Ra[2] - pinv[2]*Ra[1]);
    o18[13] = Rl[1] + (pinv[2]*Ra[0] - pinv[0]*Ra[2]);
    o18[14] = Rl[2] + (pinv[0]*Ra[1] - pinv[1]*Ra[0]);
    o18[15] = Ra[0];
    o18[16] = Ra[1];
    o18[17] = Ra[2];

    v2f* os = (v2f*)(g_out + (size_t)item * 18);
    #pragma unroll
    for (int j = 0; j < 9; ++j) os[j] = v2f{o18[2*j], o18[2*j+1]};

    v2f* od = (v2f*)(g_out + (size_t)K_TOT * 30 + (size_t)item * 6);
    #pragma unroll
    for (int j = 0; j < 3; ++j) od[j] = v2f{dv[2*j], dv[2*j+1]};
  }
}

extern "C" void kernel_launch(void* const* d_in, const int* in_sizes, int n_in,
                              void* d_out, int out_size, void* d_ws, size_t ws_size,
                              hipStream_t stream) {
  (void)in_sizes; (void)n_in; (void)out_size; (void)d_ws; (void)ws_size;
  const float* s   = (const float*)d_in[0];
  const float* a   = (const float*)d_in[1];
  const float* h0  = (const float*)d_in[2];
  const float* Wih = (const float*)d_in[3];
  const float* Whh = (const float*)d_in[4];
  const float* W1  = (const float*)d_in[5];
  const float* W2  = (const float*)d_in[6];
  const float* W3  = (const float*)d_in[7];
  float* out = (float*)d_out;

  dim3 grid(K_TOT / 128), block(128);
  hipLaunchKernelGGL(auv_step_kernel, grid, block, 0, stream,
                     s, a, h0, Wih, Whh, W1, W2, W3, out);
}